// BasicBlockTransformerConvLayer_74045236183751
// MI455X (gfx1250) — compile-verified
//
#include <hip/hip_runtime.h>
#include <math.h>

typedef __attribute__((ext_vector_type(2))) float v2f;
typedef __attribute__((ext_vector_type(8))) float v8f;

#define D 128
#define LDSS 132   // padded LDS row stride (floats): banks (4*lane+k)%64 -> conflict-free

__device__ __forceinline__ float atomicMaxFloat(float* addr, float v) {
    // signed-int max for non-negative floats, unsigned-int min for negative floats
    if (v >= 0.0f)
        return __int_as_float(atomicMax((int*)addr, __float_as_int(v)));
    else
        return __uint_as_float(atomicMin((unsigned int*)addr, __float_as_uint(v)));
}

// Fused GEMM: Q = x@Wq+bq, K = x@Wk+bk, V = x@Wv+bv, S = x@Ws+bs (S -> d_out).
// One block = one 32-row M-tile; 8 waves each own a 16-column slice of all 4 outputs.
// Each B-fragment (W slice) is reused by two WMMAs (two 16-row accumulators).
__global__ __launch_bounds__(256)
void gemm_qkvs_wmma(const float* __restrict__ x,
                    const float* __restrict__ Wq, const float* __restrict__ bq,
                    const float* __restrict__ Wk, const float* __restrict__ bk,
                    const float* __restrict__ Wv, const float* __restrict__ bv,
                    const float* __restrict__ Ws, const float* __restrict__ bs,
                    float* __restrict__ Qo, float* __restrict__ Ko,
                    float* __restrict__ Vo, float* __restrict__ So,
                    int N)
{
    __shared__ float ldsX[32 * LDSS];
    const int row0 = blockIdx.x * 32;
    const int tid  = threadIdx.x;

    // Cooperative load of the 32x128 x-tile: 1024 float4s, 4 per thread.
    #pragma unroll
    for (int i = 0; i < 4; ++i) {
        int j = tid + i * 256;       // float4 index 0..1023
        int r = j >> 5;              // row within tile
        int c = (j & 31) * 4;        // column
        float4 val;
        int grow = row0 + r;
        if (grow < N) val = *(const float4*)(x + (size_t)grow * D + c);
        else          val = make_float4(0.f, 0.f, 0.f, 0.f);
        *(float4*)(&ldsX[r * LDSS + c]) = val;   // 16B-aligned: (132r+c) % 4 == 0
    }
    __syncthreads();

    const int wave  = tid >> 5;
    const int lane  = tid & 31;
    const int lhalf = lane >> 4;     // 0 or 1
    const int lmod  = lane & 15;
    const int n0    = wave * 16;     // this wave's output-column tile

    const float* Wmat[4] = {Wq, Wk, Wv, Ws};
    const float* Bvec[4] = {bq, bk, bv, bs};
    float*       Omat[4] = {Qo, Ko, Vo, So};

    #pragma unroll
    for (int mat = 0; mat < 4; ++mat) {
        const float* __restrict__ W = Wmat[mat];
        const float bb = Bvec[mat][n0 + lmod];
        v8f acc0 = {bb, bb, bb, bb, bb, bb, bb, bb};   // rows row0 +  0..15
        v8f acc1 = {bb, bb, bb, bb, bb, bb, bb, bb};   // rows row0 + 16..31

        #pragma unroll
        for (int ks = 0; ks < 32; ++ks) {
            const int k = ks * 4 + 2 * lhalf;          // this half-wave's K pair
            v2f a0, a1, b;
            // A-frag (16x4 f32): lane<16 -> X[M=lmod][k,k+1]; lane>=16 -> K=k+2,k+3
            const float* ap0 = &ldsX[lmod * LDSS + k];
            const float* ap1 = &ldsX[(16 + lmod) * LDSS + k];
            a0[0] = ap0[0]; a0[1] = ap0[1];
            a1[0] = ap1[0]; a1[1] = ap1[1];
            // B-frag (4x16 f32): lane indexes N, (VGPR,half) index K = 2h+v
            const float* wp = W + (size_t)k * D + n0 + lmod;
            b[0] = wp[0]; b[1] = wp[D];
            acc0 = __builtin_amdgcn_wmma_f32_16x16x4_f32(false, a0, false, b,
                                                         (short)0, acc0, false, false);
            acc1 = __builtin_amdgcn_wmma_f32_16x16x4_f32(false, a1, false, b,
                                                         (short)0, acc1, false, false);
        }

        float* __restrict__ out = Omat[mat];
        #pragma unroll
        for (int r = 0; r < 8; ++r) {                  // C/D: VGPR r -> M = r + 8*lhalf
            int rowA = row0 + r + 8 * lhalf;
            int rowB = rowA + 16;
            if (rowA < N) out[(size_t)rowA * D + n0 + lmod] = acc0[r];
            if (rowB < N) out[(size_t)rowB * D + n0 + lmod] = acc1[r];
        }
    }
}

__global__ __launch_bounds__(256)
void init_md(float* __restrict__ m, float* __restrict__ den, int N)
{
    int i = blockIdx.x * blockDim.x + threadIdx.x;
    if (i < N) { m[i] = -INFINITY; den[i] = 0.0f; }
}

// One wave per edge: logits[e] = scale * dot(q[dst], k[src]); segment max into m[dst].
__global__ __launch_bounds__(256)
void edge_logits(const float* __restrict__ Q, const float* __restrict__ K,
                 const int* __restrict__ ei, float* __restrict__ logits,
                 float* __restrict__ m, int E, float scale)
{
    const int wave = threadIdx.x >> 5;
    const int lane = threadIdx.x & 31;
    const int e = blockIdx.x * 8 + wave;
    if (e >= E) return;
    const int src = ei[e];
    const int dst = ei[E + e];
    float4 qv = ((const float4*)(Q + (size_t)dst * D))[lane];
    float4 kv = ((const float4*)(K + (size_t)src * D))[lane];
    float d = qv.x * kv.x + qv.y * kv.y + qv.z * kv.z + qv.w * kv.w;
    #pragma unroll
    for (int off = 16; off; off >>= 1) d += __shfl_xor(d, off, 32);
    if (lane == 0) {
        d *= scale;
        logits[e] = d;
        atomicMaxFloat(&m[dst], d);
    }
}

// ex[e] = exp(logit - m[dst]); denom[dst] += ex[e]
__global__ __launch_bounds__(256)
void edge_exp(const float* __restrict__ logits, const int* __restrict__ ei,
              const float* __restrict__ m, float* __restrict__ ex,
              float* __restrict__ den, int E)
{
    int e = blockIdx.x * blockDim.x + threadIdx.x;
    if (e >= E) return;
    int dst = ei[E + e];
    float v = expf(logits[e] - m[dst]);
    ex[e] = v;
    atomicAdd(&den[dst], v);
}

// One wave per edge: out[dst] += (ex[e]/denom[dst]) * v[src]
__global__ __launch_bounds__(256)
void edge_scatter(const float* __restrict__ V, const int* __restrict__ ei,
                  const float* __restrict__ ex, const float* __restrict__ den,
                  float* __restrict__ out, int E)
{
    const int wave = threadIdx.x >> 5;
    const int lane = threadIdx.x & 31;
    const int e = blockIdx.x * 8 + wave;
    if (e >= E) return;
    const int src = ei[e];
    const int dst = ei[E + e];
    const float alpha = ex[e] / den[dst];
    float4 vv = ((const float4*)(V + (size_t)src * D))[lane];
    float* orow = out + (size_t)dst * D + lane * 4;
    atomicAdd(orow + 0, alpha * vv.x);
    atomicAdd(orow + 1, alpha * vv.y);
    atomicAdd(orow + 2, alpha * vv.z);
    atomicAdd(orow + 3, alpha * vv.w);
}

__global__ __launch_bounds__(256)
void elu_inplace(float* __restrict__ out, size_t n)
{
    size_t i = (size_t)blockIdx.x * blockDim.x + threadIdx.x;
    if (i < n) {
        float v = out[i];
        out[i] = v > 0.0f ? v : expm1f(v);
    }
}

extern "C" void kernel_launch(void* const* d_in, const int* in_sizes, int n_in,
                              void* d_out, int out_size, void* d_ws, size_t ws_size,
                              hipStream_t stream)
{
    const float* x  = (const float*)d_in[0];
    const int*   ei = (const int*)  d_in[1];
    const float* Wq = (const float*)d_in[2];  const float* bq = (const float*)d_in[3];
    const float* Wk = (const float*)d_in[4];  const float* bk = (const float*)d_in[5];
    const float* Wv = (const float*)d_in[6];  const float* bv = (const float*)d_in[7];
    const float* Ws = (const float*)d_in[8];  const float* bs = (const float*)d_in[9];
    float* out = (float*)d_out;

    const int N = in_sizes[0] / D;
    const int E = in_sizes[1] / 2;

    // workspace layout (floats): Q | K | V | logits | ex | m | denom
    float* ws     = (float*)d_ws;
    float* Q      = ws;
    float* K      = Q + (size_t)N * D;
    float* V      = K + (size_t)N * D;
    float* logits = V + (size_t)N * D;
    float* ex     = logits + E;
    float* m      = ex + E;
    float* den    = m + N;

    const float scale = 1.0f / sqrtf((float)D);

    const int mTiles = (N + 31) / 32;
    gemm_qkvs_wmma<<<mTiles, 256, 0, stream>>>(x, Wq, bq, Wk, bk, Wv, bv, Ws, bs,
                                               Q, K, V, out, N);
    init_md<<<(N + 255) / 256, 256, 0, stream>>>(m, den, N);
    edge_logits<<<(E + 7) / 8, 256, 0, stream>>>(Q, K, ei, logits, m, E, scale);
    edge_exp<<<(E + 255) / 256, 256, 0, stream>>>(logits, ei, m, ex, den, E);
    edge_scatter<<<(E + 7) / 8, 256, 0, stream>>>(V, ei, ex, den, out, E);
    const size_t tot = (size_t)N * D;
    elu_inplace<<<(tot + 255) / 256, 256, 0, stream>>>(out, tot);
}